// LocalizingMask_28784870817858
// MI455X (gfx1250) — compile-verified
//
#include <hip/hip_runtime.h>
#include <stdint.h>

// Problem constants (baked into the reference)
#define NDIM 8192
#define THRESH 0.5f

// Tiling
#define LANES 32
#define WAVES 8
#define BLOCK_THREADS (LANES * WAVES)                 // 256 threads = 8 wave32
#define COLS_PER_WAVE 128                             // 32 lanes * float4
#define COLS_PER_BLOCK (COLS_PER_WAVE * WAVES)        // 1024
#define ROWS_PER_TILE 8
#define TILES_PER_BLOCK 4
#define ROWS_PER_BLOCK (ROWS_PER_TILE * TILES_PER_BLOCK) // 32
#define TILE_FLOATS (ROWS_PER_TILE * COLS_PER_WAVE)   // 1024 floats
#define TILE_BYTES (TILE_FLOATS * 4)                  // 4 KB per tile buffer

typedef float f32x4 __attribute__((ext_vector_type(4)));
typedef unsigned int u32x4 __attribute__((ext_vector_type(4)));
typedef int i32x8 __attribute__((ext_vector_type(8)));
typedef int i32x4 __attribute__((ext_vector_type(4)));

// Issue one TDM DMA: 2D tile (ROWS_PER_TILE x COLS_PER_WAVE f32, row stride NDIM)
// from global memory into LDS at byte offset lds_byte_off. Tracked by TENSORcnt.
// D# layout per CDNA5 ISA §8 (group0 128b, group1 256b; groups 2/3 unused for 2D).
__device__ __forceinline__ void tdm_load_tile(uint32_t lds_byte_off, const float* gsrc) {
  asm volatile("" ::: "memory"); // keep prior LDS reads before buffer reuse
  uint64_t ga = (uint64_t)(uintptr_t)gsrc;
  u32x4 g0;
  g0[0] = 1u;                                    // count=1, user descriptor
  g0[1] = lds_byte_off;                          // D#.lds_addr (bytes)
  g0[2] = (uint32_t)ga;                          // global_addr[31:0]
  g0[3] = (uint32_t)((ga >> 32) & 0x01FFFFFFu)   // global_addr[56:32]
          | (2u << 30);                          // type=2 ("image"/tensor)
  i32x8 g1;
  g1[0] = (int)(2u << 16);                          // workgroup_mask=0, data_size=2 (4B)
  g1[1] = (int)((uint32_t)COLS_PER_WAVE << 16);     // tensor_dim0[15:0] @ bits 63:48
  g1[2] = (int)((uint32_t)ROWS_PER_TILE << 16);     // tensor_dim0 hi=0 | tensor_dim1 lo
  g1[3] = (int)((uint32_t)COLS_PER_WAVE << 16);     // tensor_dim1 hi=0 | tile_dim0=128
  g1[4] = (int)(uint32_t)ROWS_PER_TILE;             // tile_dim1=8 | tile_dim2=0
  g1[5] = (int)(uint32_t)NDIM;                      // tensor_dim0_stride[31:0] = 8192
  g1[6] = 0;                                        // stride0 hi | stride1 lo
  g1[7] = 0;                                        // stride1 hi
  i32x4 gz4 = {0, 0, 0, 0};                         // group2 (unused, 2D tile)
  i32x8 gz8 = {0, 0, 0, 0, 0, 0, 0, 0};             // group3/extra (unused)
  // 6-arg toolchain form: (g0, g1, g2, g3, g4, cpol)
  __builtin_amdgcn_tensor_load_to_lds(g0, g1, gz4, gz4, gz8, 0);
}

__global__ __launch_bounds__(BLOCK_THREADS)
void localizing_mask_kernel(const float* __restrict__ L,
                            const float* __restrict__ f,
                            float* __restrict__ out) {
  // Per-wave double-buffered tile staging: 8 waves * 2 * 4KB = 64 KB LDS
  __shared__ __align__(16) float lds[WAVES * 2 * TILE_FLOATS];

  const int wid  = __builtin_amdgcn_readfirstlane((int)(threadIdx.x >> 5));
  const int lane = (int)(threadIdx.x & 31u);
  const int colTile = blockIdx.x * COLS_PER_BLOCK + wid * COLS_PER_WAVE;
  const int col     = colTile + lane * 4;
  const int rowBase = blockIdx.y * ROWS_PER_BLOCK;

  // f is column-indexed; columns are fixed per lane -> load once, reuse for all rows.
  const f32x4 fv = *(const f32x4*)(f + col);

  const uint32_t ldsWaveByte =
      (uint32_t)(uintptr_t)(void*)lds + (uint32_t)wid * 2u * TILE_BYTES;
  const float* ldsWave = lds + wid * 2 * TILE_FLOATS;

  // Prologue: DMA tile 0 into buffer 0.
  tdm_load_tile(ldsWaveByte, L + (size_t)rowBase * NDIM + colTile);

#pragma unroll
  for (int t = 0; t < TILES_PER_BLOCK; ++t) {
    if (t + 1 < TILES_PER_BLOCK) {
      // Prefetch next tile into the other buffer, then wait for current (in-order).
      tdm_load_tile(ldsWaveByte + (uint32_t)((t + 1) & 1) * TILE_BYTES,
                    L + (size_t)(rowBase + (t + 1) * ROWS_PER_TILE) * NDIM + colTile);
      __builtin_amdgcn_s_wait_tensorcnt(1);
    } else {
      __builtin_amdgcn_s_wait_tensorcnt(0);
    }
    asm volatile("" ::: "memory"); // LDS reads below must stay below the wait

    const float* src = ldsWave + (t & 1) * TILE_FLOATS;
#pragma unroll
    for (int r = 0; r < ROWS_PER_TILE; ++r) {
      f32x4 x = *(const f32x4*)(src + r * COLS_PER_WAVE + lane * 4);
      f32x4 o;
      o.x = (fabsf(x.x * x.x * x.x * fv.x) > THRESH) ? 1.0f : 0.0f;
      o.y = (fabsf(x.y * x.y * x.y * fv.y) > THRESH) ? 1.0f : 0.0f;
      o.z = (fabsf(x.z * x.z * x.z * fv.z) > THRESH) ? 1.0f : 0.0f;
      o.w = (fabsf(x.w * x.w * x.w * fv.w) > THRESH) ? 1.0f : 0.0f;
      // Output is streamed once (256 MB > 192 MB L2): non-temporal store.
      __builtin_nontemporal_store(
          o, (f32x4*)(out + (size_t)(rowBase + t * ROWS_PER_TILE + r) * NDIM + col));
    }
  }
}

extern "C" void kernel_launch(void* const* d_in, const int* in_sizes, int n_in,
                              void* d_out, int out_size, void* d_ws, size_t ws_size,
                              hipStream_t stream) {
  (void)in_sizes; (void)n_in; (void)out_size; (void)d_ws; (void)ws_size;
  const float* L = (const float*)d_in[0];   // [8192, 8192] f32
  const float* f = (const float*)d_in[1];   // [8192] f32
  float* out = (float*)d_out;               // [8192, 8192] mask as f32 (1.0/0.0)

  dim3 grid(NDIM / COLS_PER_BLOCK, NDIM / ROWS_PER_BLOCK); // (8, 256)
  dim3 block(BLOCK_THREADS);                               // 256 = 8 wave32
  localizing_mask_kernel<<<grid, block, 0, stream>>>(L, f, out);
}